// SRGNN_41446434406577
// MI455X (gfx1250) — compile-verified
//
#include <hip/hip_runtime.h>

// SR-GNN forward for MI455X (gfx1250, wave32, WMMA), round 3: fully fused.
// B=4096, L=50 (padded 64), D=64, V=100001, fp32 end-to-end.
//
// Single kernel, one block per batch element. Traffic ~240MB (A 82 + emb
// gather 52 + pos/neg emb 104 + out 1.6) -> ~10us at 23.3TB/s. All GEMMs via
// V_WMMA_F32_16X16X4_F32; global gathers via GLOBAL_LOAD_ASYNC_TO_LDS_B128.
//
// Dynamic LDS (64KB): H[64*64] hidden (pre- then post-GRU) |
//                     W[64*64] h_in/h_out, later seq_hidden | I[64*128] inputs

#define BATCH 4096
#define LSEQ  50
#define DDIM  64
#define LPAD  64

typedef __attribute__((ext_vector_type(2))) float v2f;
typedef __attribute__((ext_vector_type(8))) float v8f;

static __device__ __forceinline__ v8f wmma_k4(v2f a, v2f b, v8f c) {
  return __builtin_amdgcn_wmma_f32_16x16x4_f32(false, a, false, b, (short)0, c,
                                               false, false);
}

// A-fragment: 16x4 fp32. lane&15 = M; lanes 0-15: K={k0,k0+1}, 16-31: K={k0+2,k0+3}.
static __device__ __forceinline__ v2f ldA(const float* M, int lda, int row0,
                                          int k0, int lane) {
  int m = lane & 15;
  int kk = (lane >> 4) << 1;
  const float* p = M + (row0 + m) * lda + (k0 + kk);
  v2f r; r.x = p[0]; r.y = p[1];
  return r;
}
// B-fragment: 4x16 fp32 row-major.
static __device__ __forceinline__ v2f ldB(const float* M, int ldb, int k0,
                                          int col0, int lane) {
  int n = lane & 15;
  int kk = (lane >> 4) << 1;
  v2f r;
  r.x = M[(k0 + kk) * ldb + col0 + n];
  r.y = M[(k0 + kk + 1) * ldb + col0 + n];
  return r;
}
// B-fragment from transposed weight: B[k][n] = W[n][k], W row-major (N x K).
static __device__ __forceinline__ v2f ldBT(const float* W, int ldw, int k0,
                                           int col0, int lane) {
  int n = lane & 15;
  int kk = (lane >> 4) << 1;
  const float* p = W + (col0 + n) * ldw + (k0 + kk);
  v2f r; r.x = p[0]; r.y = p[1];
  return r;
}
// A-operand from global adjacency slice with bounds masking (i<50, j<50).
static __device__ __forceinline__ v2f ldA_adj(const float* Ab, int row0, int k0,
                                              int lane) {
  int i = row0 + (lane & 15);
  int j = k0 + ((lane >> 4) << 1);
  v2f r; r.x = 0.f; r.y = 0.f;
  if (i < LSEQ) {
    const float* p = Ab + i * (2 * LSEQ) + j;
    if (j < LSEQ) r.x = p[0];
    if (j + 1 < LSEQ) r.y = p[1];
  }
  return r;
}

static __device__ __forceinline__ float sigmoidf(float x) {
  return 1.0f / (1.0f + __expf(-x));
}

// LDS byte offset = low 32 bits of the generic shared-aperture address (ISA 10.2).
static __device__ __forceinline__ unsigned lds_off(const void* p) {
  return (unsigned)(unsigned long long)p;
}
// Async 16B memory->LDS copy (per-lane addresses), tracked by ASYNCcnt.
static __device__ __forceinline__ void async_g2l_b128(const float* gsrc,
                                                      const float* ldst) {
  asm volatile("global_load_async_to_lds_b128 %0, %1, off"
               :: "v"(lds_off(ldst)), "v"(gsrc)
               : "memory");
}
static __device__ __forceinline__ void wait_async0() {
  asm volatile("s_wait_asynccnt 0x0" ::: "memory");
}

// ================= Fully fused SR-GNN forward =============================
__global__ void srgnn_fused(const int* __restrict__ item_seq,
                            const int* __restrict__ alias_inputs,
                            const int* __restrict__ items,
                            const int* __restrict__ pos_seqs,
                            const int* __restrict__ neg_seqs,
                            const float* __restrict__ A,
                            const float* __restrict__ emb,
                            const float* __restrict__ we_in_w,
                            const float* __restrict__ we_in_b,
                            const float* __restrict__ we_out_w,
                            const float* __restrict__ we_out_b,
                            const float* __restrict__ b_iah,
                            const float* __restrict__ b_ioh,
                            const float* __restrict__ w_ih,
                            const float* __restrict__ b_ih,
                            const float* __restrict__ w_hh,
                            const float* __restrict__ b_hh,
                            const float* __restrict__ lin1_w,
                            const float* __restrict__ lin1_b,
                            const float* __restrict__ lin2_w,
                            const float* __restrict__ lin2_b,
                            const float* __restrict__ lin3_w,
                            const float* __restrict__ lintr_w,
                            const float* __restrict__ lintr_b,
                            float* __restrict__ out) {
  extern __shared__ float smem[];
  float* H = smem;            // 4096 floats: hidden (pre-GRU, then post-GRU)
  float* W = H + LPAD * DDIM; // 4096 floats: h_in/h_out, then seq_hidden
  float* I = W + LPAD * DDIM; // 8192 floats: GRU inputs
  __shared__ float q1s[DDIM], alphas[LPAD], ht_s[DDIM], avec[DDIM], so_s[DDIM];

  int b = blockIdx.x;
  int tid = threadIdx.x;
  int wave = tid >> 5, lane = tid & 31;
  int n = lane & 15, mb = (lane >> 4) << 3;

  // ---- gather emb[items] -> H (async 16B/lane; zero-pad rows >= LSEQ) ----
  for (int idx = tid * 4; idx < LPAD * DDIM; idx += 1024) {
    int l = idx >> 6, k = idx & 63;
    if (l < LSEQ) {
      const float* g = emb + (long)items[b * LSEQ + l] * DDIM + k;
      async_g2l_b128(g, H + idx);
    } else {
      H[idx] = 0.f; H[idx + 1] = 0.f; H[idx + 2] = 0.f; H[idx + 3] = 0.f;
    }
  }
  wait_async0();
  __syncthreads();

  const float* Abase = A + (long)b * LSEQ * 2 * LSEQ;

  // ---- graph propagation: I = [A_in @ (H@we_in^T+b) + b_iah |
  //                              A_out @ (H@we_out^T+b) + b_ioh] ----
  for (int phase = 0; phase < 2; phase++) {
    const float* Ww = phase ? we_out_w : we_in_w;
    const float* Wb = phase ? we_out_b : we_in_b;
    const float* gb = phase ? b_ioh : b_iah;
    const float* Ap = Abase + (phase ? LSEQ : 0);

    for (int tt = wave; tt < 16; tt += 8) {     // W = H @ Ww^T + Wb
      int mt = tt >> 2, nt = tt & 3;
      v8f c = {};
#pragma unroll
      for (int k0 = 0; k0 < DDIM; k0 += 4) {
        v2f a = ldA(H, DDIM, mt * 16, k0, lane);
        v2f bf = ldBT(Ww, DDIM, k0, nt * 16, lane);
        c = wmma_k4(a, bf, c);
      }
      int col = nt * 16 + n;
      float bv = Wb[col];
#pragma unroll
      for (int v = 0; v < 8; v++)
        W[(mt * 16 + mb + v) * DDIM + col] = c[v] + bv;
    }
    __syncthreads();

    for (int tt = wave; tt < 16; tt += 8) {     // I half = A_slice @ W + gb
      int mt = tt >> 2, nt = tt & 3;
      v8f c = {};
#pragma unroll
      for (int k0 = 0; k0 < 52; k0 += 4) {      // K padded; j>=50 masked to 0
        v2f a = ldA_adj(Ap, mt * 16, k0, lane);
        v2f bf = ldB(W, DDIM, k0, nt * 16, lane);
        c = wmma_k4(a, bf, c);
      }
      int col = nt * 16 + n;
      float bv = gb[col];
#pragma unroll
      for (int v = 0; v < 8; v++)
        I[(mt * 16 + mb + v) * 128 + phase * DDIM + col] = c[v] + bv;
    }
    __syncthreads();
  }

  // ---- GRU: gi = I @ w_ih^T, gh = H @ w_hh^T; outputs kept in registers ---
  // wave owns M-tile (wave&3) and t-tiles {2*(wave>>2), 2*(wave>>2)+1}
  int mt = wave & 3;
  int tbase = (wave >> 2) << 1;
  const float* Irow = I + mt * 16 * 128;
  const float* Hrow = H + mt * 16 * DDIM;
  float hnew[2][8];
#pragma unroll
  for (int ti = 0; ti < 2; ti++) {
    int t = tbase + ti;
    v8f gr = {}, gz = {}, gn = {}, hr = {}, hz = {}, hn = {};
    for (int k0 = 0; k0 < 128; k0 += 4) {
      v2f a = ldA(Irow, 128, 0, k0, lane);
      gr = wmma_k4(a, ldBT(w_ih, 128, k0, t * 16, lane), gr);
      gz = wmma_k4(a, ldBT(w_ih, 128, k0, 64 + t * 16, lane), gz);
      gn = wmma_k4(a, ldBT(w_ih, 128, k0, 128 + t * 16, lane), gn);
    }
    for (int k0 = 0; k0 < 64; k0 += 4) {
      v2f a = ldA(Hrow, DDIM, 0, k0, lane);
      hr = wmma_k4(a, ldBT(w_hh, 64, k0, t * 16, lane), hr);
      hz = wmma_k4(a, ldBT(w_hh, 64, k0, 64 + t * 16, lane), hz);
      hn = wmma_k4(a, ldBT(w_hh, 64, k0, 128 + t * 16, lane), hn);
    }
    int col = t * 16 + n;
    float bir = b_ih[col], biz = b_ih[64 + col], bin_ = b_ih[128 + col];
    float bhr = b_hh[col], bhz = b_hh[64 + col], bhn = b_hh[128 + col];
#pragma unroll
    for (int v = 0; v < 8; v++) {
      float hv = Hrow[(mb + v) * DDIM + col];
      float rg = sigmoidf(gr[v] + bir + hr[v] + bhr);
      float ig = sigmoidf(gz[v] + biz + hz[v] + bhz);
      float ng = tanhf(gn[v] + bin_ + rg * (hn[v] + bhn));
      hnew[ti][v] = (1.f - ig) * hv + ig * ng;
    }
  }
  __syncthreads();   // all waves finished reading H
#pragma unroll
  for (int ti = 0; ti < 2; ti++) {
    int col = (tbase + ti) * 16 + n;
#pragma unroll
    for (int v = 0; v < 8; v++)
      H[(mt * 16 + mb + v) * DDIM + col] = hnew[ti][v];   // H := post-GRU hidden
  }
  if (tid < LPAD) alphas[tid] = 0.f;
  __syncthreads();

  // ---- seq_hidden = H[alias] -> W (LDS permute); ht = seq_hidden[49] ----
  for (int idx = tid; idx < LPAD * DDIM; idx += 256) {
    int l = idx >> 6, k = idx & 63;
    W[idx] = (l < LSEQ) ? H[alias_inputs[b * LSEQ + l] * DDIM + k] : 0.f;
  }
  __syncthreads();
  if (tid < DDIM) ht_s[tid] = W[(LSEQ - 1) * DDIM + tid];
  __syncthreads();
  if (tid < DDIM) {                 // q1 = ht @ lin1^T + lin1_b
    float s = lin1_b[tid];
    const float* w = lin1_w + tid * DDIM;
    for (int k = 0; k < DDIM; k++) s += ht_s[k] * w[k];
    q1s[tid] = s;
  }
  __syncthreads();

  // ---- q2 = seq_hidden @ lin2^T (WMMA); alpha = sigmoid(q1+q2) @ lin3^T ---
  for (int tt = wave; tt < 16; tt += 8) {
    int mt2 = tt >> 2, nt = tt & 3;
    v8f c = {};
#pragma unroll
    for (int k0 = 0; k0 < DDIM; k0 += 4) {
      v2f a = ldA(W, DDIM, mt2 * 16, k0, lane);
      v2f bf = ldBT(lin2_w, DDIM, k0, nt * 16, lane);
      c = wmma_k4(a, bf, c);
    }
    int col = nt * 16 + n;
    float add = lin2_b[col] + q1s[col];
    float w3 = lin3_w[col];
#pragma unroll
    for (int v = 0; v < 8; v++) {
      float s = sigmoidf(c[v] + add) * w3;
      atomicAdd(&alphas[mt2 * 16 + mb + v], s);
    }
  }
  __syncthreads();
  if (tid < LPAD) {                 // apply mask
    float m = (tid < LSEQ && item_seq[b * LSEQ + tid] > 0) ? 1.f : 0.f;
    alphas[tid] *= m;
  }
  __syncthreads();
  if (tid < DDIM) {                 // a = sum_l alpha_l * seq_hidden_l
    float s = 0.f;
    for (int l = 0; l < LSEQ; l++) s += alphas[l] * W[l * DDIM + tid];
    avec[tid] = s;
  }
  __syncthreads();
  if (tid < DDIM) {                 // seq_output = [a|ht] @ lintr^T + lintr_b
    float s = lintr_b[tid];
    const float* w = lintr_w + tid * 128;
    for (int k = 0; k < DDIM; k++) s += avec[k] * w[k] + ht_s[k] * w[DDIM + k];
    so_s[tid] = s;
  }
  __syncthreads();

  // ---- scores: lane holds 2 of 64 dims; shuffle-reduce across the wave ----
  float s0 = so_s[lane * 2], s1 = so_s[lane * 2 + 1];
  for (int l = wave; l < LSEQ; l += 8) {
    const float* pe = emb + (long)pos_seqs[b * LSEQ + l] * DDIM;
    const float* ne = emb + (long)neg_seqs[b * LSEQ + l] * DDIM;
    if (l + 8 < LSEQ) {
      __builtin_prefetch(emb + (long)pos_seqs[b * LSEQ + l + 8] * DDIM, 0, 0);
      __builtin_prefetch(emb + (long)neg_seqs[b * LSEQ + l + 8] * DDIM, 0, 0);
    }
    float p = s0 * pe[lane * 2] + s1 * pe[lane * 2 + 1];
    float q = s0 * ne[lane * 2] + s1 * ne[lane * 2 + 1];
#pragma unroll
    for (int off = 16; off > 0; off >>= 1) {
      p += __shfl_down(p, off, 32);
      q += __shfl_down(q, off, 32);
    }
    if (lane == 0) {
      out[b * LSEQ + l] = p;                  // pos_score
      out[BATCH * LSEQ + b * LSEQ + l] = q;   // neg_score
    }
  }
}

extern "C" void kernel_launch(void* const* d_in, const int* in_sizes, int n_in,
                              void* d_out, int out_size, void* d_ws,
                              size_t ws_size, hipStream_t stream) {
  (void)in_sizes; (void)n_in; (void)out_size; (void)d_ws; (void)ws_size;
  const int* item_seq     = (const int*)d_in[0];
  const int* alias_inputs = (const int*)d_in[1];
  const int* items        = (const int*)d_in[2];
  const int* pos_seqs     = (const int*)d_in[3];
  const int* neg_seqs     = (const int*)d_in[4];
  const float* A        = (const float*)d_in[5];
  const float* emb      = (const float*)d_in[6];
  const float* we_in_w  = (const float*)d_in[7];
  const float* we_in_b  = (const float*)d_in[8];
  const float* we_out_w = (const float*)d_in[9];
  const float* we_out_b = (const float*)d_in[10];
  const float* b_iah    = (const float*)d_in[11];
  const float* b_ioh    = (const float*)d_in[12];
  const float* w_ih     = (const float*)d_in[13];
  const float* b_ih     = (const float*)d_in[14];
  const float* w_hh     = (const float*)d_in[15];
  const float* b_hh     = (const float*)d_in[16];
  const float* lin1_w   = (const float*)d_in[17];
  const float* lin1_b   = (const float*)d_in[18];
  const float* lin2_w   = (const float*)d_in[19];
  const float* lin2_b   = (const float*)d_in[20];
  const float* lin3_w   = (const float*)d_in[21];
  const float* lintr_w  = (const float*)d_in[22];
  const float* lintr_b  = (const float*)d_in[23];
  float* out = (float*)d_out;

  const size_t lds_bytes = (size_t)(LPAD * DDIM + LPAD * DDIM + LPAD * 128) *
                           sizeof(float);   // 64 KB dynamic LDS
  srgnn_fused<<<BATCH, 256, lds_bytes, stream>>>(
      item_seq, alias_inputs, items, pos_seqs, neg_seqs, A, emb, we_in_w,
      we_in_b, we_out_w, we_out_b, b_iah, b_ioh, w_ih, b_ih, w_hh, b_hh,
      lin1_w, lin1_b, lin2_w, lin2_b, lin3_w, lintr_w, lintr_b, out);
}